// EntmaxBisectLoss_83923660964517
// MI455X (gfx1250) — compile-verified
//
#include <hip/hip_runtime.h>
#include <hip/hip_bf16.h>
#include <stdint.h>

// EntmaxBisectLoss for MI455X (gfx1250).
// One 1024-thread (32 x wave32) workgroup per row. The 128KB row is staged
// into LDS once via gfx1250 async global->LDS b128 loads (ASYNCcnt path,
// confirmed in disasm), each thread keeps its 32 strided elements in
// registers, and the 50 bisection reductions run register->butterfly->LDS
// with one barrier each. HBM is read exactly once (~512MB -> ~22us at
// 23.3TB/s); the rest is VALU-bound (~2e10 FMA-class ops).

#define D            32000
#define NROWS        4096
#define T            1024
#define NWAVES       32
#define EPT          32      // ceil(D / T)
#define VEC4         8000    // D / 4
#define N_ITER_      50
#define IGNORE_INDEX (-100)

__device__ __forceinline__ float wave_sum(float v) {
#pragma unroll
  for (int off = 16; off > 0; off >>= 1) v += __shfl_xor(v, off, 32);
  return v;
}

__device__ __forceinline__ float wave_max(float v) {
#pragma unroll
  for (int off = 16; off > 0; off >>= 1) v = fmaxf(v, __shfl_xor(v, off, 32));
  return v;
}

// Two-level block reduction (NWAVES == 32 == wave width).
// One barrier. Every thread returns the identical block total (same butterfly
// order in every wave -> bitwise deterministic). Caller alternates between
// two LDS buffers so the next call's writes can't race this call's reads.
__device__ __forceinline__ float block_sum(float v, float* buf, int lane, int wave) {
  v = wave_sum(v);
  if (lane == 0) buf[wave] = v;
  __syncthreads();
  float t = buf[lane];          // lanes 0..31 pick up the 32 wave partials
  return wave_sum(t);
}

__device__ __forceinline__ float block_max(float v, float* buf, int lane, int wave) {
  v = wave_max(v);
  if (lane == 0) buf[wave] = v;
  __syncthreads();
  float t = buf[lane];
  return wave_max(t);
}

__global__ __launch_bounds__(T) void entmax_row_kernel(
    const float* __restrict__ X, const int* __restrict__ target,
    float* __restrict__ row_loss, float* __restrict__ row_valid) {
  __shared__ float lds_x[D];           // raw X row (128000 B)
  __shared__ float red[2][NWAVES];     // double-buffered reduction scratch

  const int tid  = (int)threadIdx.x;
  const int lane = tid & 31;
  const int wave = tid >> 5;
  const int row  = (int)blockIdx.x;
  const float* rowX = X + (size_t)row * D;   // uniform -> SGPR pair

  // ---- stage row into LDS via gfx1250 async global->LDS (B128, GVS mode) ----
  {
    unsigned lds_base = (unsigned)(uintptr_t)&lds_x[0];
#pragma unroll
    for (int k = 0; k < 8; ++k) {
      int j = tid + k * T;                     // float4 index
      if (j < VEC4) {
        unsigned voff  = (unsigned)j * 16u;    // byte offset within row
        unsigned laddr = lds_base + voff;      // LDS byte address
        asm volatile("global_load_async_to_lds_b128 %0, %1, %2"
                     :: "v"(laddr), "v"(voff), "s"(rowX)
                     : "memory");
      }
    }
    asm volatile("s_wait_asynccnt 0x0" ::: "memory");
  }
  __syncthreads();

  // ---- pull this thread's 32 strided elements into registers as Xs = 0.5*X ----
  // (stride-T b32 reads: consecutive lanes hit consecutive banks, conflict-free)
  float xs[EPT];
#pragma unroll
  for (int k = 0; k < EPT; ++k) {
    int i = tid + k * T;
    xs[k] = (i < D) ? 0.5f * lds_x[i] : -3.0e38f;   // pad -> relu()==0 everywhere
  }

  int b = 0;

  // ---- bracket ----
  float m = -3.0e38f;
#pragma unroll
  for (int k = 0; k < EPT; ++k) m = fmaxf(m, xs[k]);
  float max_val = block_max(m, red[b], lane, wave); b ^= 1;

  float tau_lo = max_val - 1.0f;
  float tau_hi = max_val - 0.005590169943749474f;   // (1/32000)^(alpha-1), alpha=1.5

  float acc = 0.0f;
#pragma unroll
  for (int k = 0; k < EPT; ++k) {
    float t = fmaxf(xs[k] - tau_lo, 0.0f);
    acc = fmaf(t, t, acc);
  }
  float f_lo = block_sum(acc, red[b], lane, wave) - 1.0f; b ^= 1;

  // ---- 50 bisection steps: one register pass + one barrier each ----
  float dm    = tau_hi - tau_lo;
  float tau_m = tau_lo;
  for (int it = 0; it < N_ITER_; ++it) {
    dm *= 0.5f;
    tau_m = tau_lo + dm;
    acc = 0.0f;
#pragma unroll
    for (int k = 0; k < EPT; ++k) {
      float t = fmaxf(xs[k] - tau_m, 0.0f);
      acc = fmaf(t, t, acc);
    }
    float f_m = block_sum(acc, red[b], lane, wave) - 1.0f; b ^= 1;
    if (f_m * f_lo >= 0.0f) tau_lo = tau_m;   // identical decision in every thread
  }

  // ---- final pass: S = sum q, S3 = sum t^3 (= q^1.5), SX = sum q*Xs ----
  float a2 = 0.0f, a3 = 0.0f, ax = 0.0f;
#pragma unroll
  for (int k = 0; k < EPT; ++k) {
    float t = fmaxf(xs[k] - tau_m, 0.0f);
    float q = t * t;
    a2 += q;
    a3 = fmaf(q, t, a3);
    ax = fmaf(q, xs[k], ax);
  }
  float S  = block_sum(a2, red[b], lane, wave); b ^= 1;
  float S3 = block_sum(a3, red[b], lane, wave); b ^= 1;
  float SX = block_sum(ax, red[b], lane, wave); b ^= 1;

  if (tid == 0) {
    int  traw  = target[row];
    bool valid = (traw != IGNORE_INDEX);
    int  tgt   = valid ? traw : 0;
    float x_t  = lds_x[tgt];                       // raw X[row, tgt]
    // p_i = q_i / S ;  sum p^1.5 = S3 / S^1.5 ;  sum p*X = 2*SX / S
    float sum_p_alpha = S3 / (S * sqrtf(S));
    float omega = (1.0f - sum_p_alpha) * (1.0f / 0.75f);   // /(alpha*(alpha-1))
    float dot   = (2.0f * SX) / S - x_t;
    float loss  = valid ? (omega + dot) : 0.0f;
    row_loss[row]  = loss;
    row_valid[row] = valid ? 1.0f : 0.0f;
  }
}

__global__ __launch_bounds__(T) void finalize_kernel(
    const float* __restrict__ row_loss, const float* __restrict__ row_valid,
    float* __restrict__ out) {
  __shared__ float sbuf[NWAVES];
  __shared__ float cbuf[NWAVES];
  const int tid = (int)threadIdx.x, lane = tid & 31, wave = tid >> 5;
  float s = 0.0f, c = 0.0f;
  for (int i = tid; i < NROWS; i += T) { s += row_loss[i]; c += row_valid[i]; }
  s = wave_sum(s);
  c = wave_sum(c);
  if (lane == 0) { sbuf[wave] = s; cbuf[wave] = c; }
  __syncthreads();
  if (tid == 0) {
    float ts = 0.0f, tc = 0.0f;
#pragma unroll
    for (int w = 0; w < NWAVES; ++w) { ts += sbuf[w]; tc += cbuf[w]; }
    out[0] = ts / fmaxf(tc, 1.0f);   // reduction='mean' over valid rows
  }
}

extern "C" void kernel_launch(void* const* d_in, const int* in_sizes, int n_in,
                              void* d_out, int out_size, void* d_ws, size_t ws_size,
                              hipStream_t stream) {
  (void)in_sizes; (void)n_in; (void)out_size; (void)ws_size;
  const float* X      = (const float*)d_in[0];
  const int*   target = (const int*)d_in[1];
  float* row_loss  = (float*)d_ws;          // 4096 floats
  float* row_valid = row_loss + NROWS;      // 4096 floats

  entmax_row_kernel<<<NROWS, T, 0, stream>>>(X, target, row_loss, row_valid);
  finalize_kernel<<<1, T, 0, stream>>>(row_loss, row_valid, (float*)d_out);
}